// SAGEConvolution_72911364817007
// MI455X (gfx1250) — compile-verified
//
#include <hip/hip_runtime.h>
#include <hip/hip_bf16.h>

// ---------------------------------------------------------------------------
// GraphSAGE 2-layer forward for MI455X (gfx1250, wave32).
// Aggregation: L2-resident scatter-add with f32 global atomics.
// GEMMs: V_WMMA_F32_16X16X4_F32 (full f32 precision; compute is far from the
// roofline so f32 WMMA is the right choice vs. lossy f16/bf16).
// ---------------------------------------------------------------------------

typedef __attribute__((ext_vector_type(2))) float v2f;
typedef __attribute__((ext_vector_type(8))) float v8f;

#define N_NODES 100000
#define N_FEAT  128

// ---------------- utility kernels ----------------

__global__ void zero_f32_kernel(float* __restrict__ p, long n) {
    long i = (long)blockIdx.x * blockDim.x + threadIdx.x;
    long stride = (long)gridDim.x * blockDim.x;
    for (; i < n; i += stride) p[i] = 0.0f;
}

__global__ void degree_kernel(const int* __restrict__ dst, float* __restrict__ deg, long n_edges) {
    long e = (long)blockIdx.x * blockDim.x + threadIdx.x;
    if (e < n_edges) atomicAdd(&deg[dst[e]], 1.0f);
}

__global__ void invdeg_kernel(float* __restrict__ deg, long n) {
    long i = (long)blockIdx.x * blockDim.x + threadIdx.x;
    if (i < n) deg[i] = 1.0f / fmaxf(deg[i], 1.0f);
}

// One wave (32 lanes) per edge; each lane moves 4 contiguous floats.
// Row gather = coalesced global_load_b128; scatter = 4x global_atomic_add_f32.
// Working set (feat + agg = ~102 MB) fits in the 192 MB L2, so atomics stay on-chip.
__global__ void scatter_add_kernel(const float* __restrict__ feat,
                                   const int* __restrict__ src,
                                   const int* __restrict__ dst,
                                   float* __restrict__ agg, long n_edges) {
    long idx  = (long)blockIdx.x * blockDim.x + threadIdx.x;
    long e    = idx >> 5;
    int  lane = (int)(idx & 31);
    if (e >= n_edges) return;
    int s = src[e];
    int d = dst[e];
    float4 v = reinterpret_cast<const float4*>(feat + (long)s * N_FEAT)[lane];
    float* dp = agg + (long)d * N_FEAT + lane * 4;
    atomicAdd(dp + 0, v.x);
    atomicAdd(dp + 1, v.y);
    atomicAdd(dp + 2, v.z);
    atomicAdd(dp + 3, v.w);
}

// ---------------- WMMA GEMM ----------------
// out[node0:node0+16, :n_out] = relu?( (agg*inv_deg) @ Wl + self_feat @ Wr + bias )
// One wave computes one 16x16 output tile using V_WMMA_F32_16X16X4_F32.
// blockDim.x = 32 * (n_out/16); grid.x = N_NODES/16 (100000 % 16 == 0, so
// EXEC is all-1s in every WMMA wave as the ISA requires).
//
// f32 16x4 A fragment (ISA 7.12.2): lanes 0-15 hold (K, K+1) of row M=lane;
// lanes 16-31 hold (K+2, K+3) of row M=lane-16. B (4x16) mirrors it.
// f32 16x16 C/D: VGPR r -> M=r (lanes 0-15) / M=r+8 (lanes 16-31), N=lane&15.
__global__ __launch_bounds__(256) void sage_gemm_wmma_kernel(
    const float* __restrict__ agg, const float* __restrict__ inv_deg,
    const float* __restrict__ self_feat,
    const float* __restrict__ Wl, const float* __restrict__ Wr,
    const float* __restrict__ bias,
    float* __restrict__ out, int n_out, int relu) {

    const int node0 = blockIdx.x * 16;
    const int lane  = threadIdx.x & 31;
    const int wave  = threadIdx.x >> 5;
    const int n0    = wave * 16;          // output column tile
    const int half  = lane >> 4;          // 0: lanes 0-15, 1: lanes 16-31
    const int l15   = lane & 15;
    const int koff  = half * 2;           // K sub-offset for this lane half

    const int   rowA = node0 + l15;
    const float id   = inv_deg[rowA];
    const float* aggRow  = agg       + (long)rowA * N_FEAT;
    const float* selfRow = self_feat + (long)rowA * N_FEAT;

    v8f acc = {};

    // Pass 1: (agg * inv_deg) @ Wl   (inv_deg scaling fused into A fragment)
    #pragma unroll 4
    for (int k = 0; k < N_FEAT; k += 4) {
        v2f a;
        a.x = aggRow[k + koff]     * id;
        a.y = aggRow[k + koff + 1] * id;
        v2f b;
        b.x = Wl[(k + koff)     * n_out + n0 + l15];
        b.y = Wl[(k + koff + 1) * n_out + n0 + l15];
        acc = __builtin_amdgcn_wmma_f32_16x16x4_f32(false, a, false, b,
                                                    (short)0, acc, false, false);
    }
    // Pass 2: self_feat @ Wr
    #pragma unroll 4
    for (int k = 0; k < N_FEAT; k += 4) {
        v2f a;
        a.x = selfRow[k + koff];
        a.y = selfRow[k + koff + 1];
        v2f b;
        b.x = Wr[(k + koff)     * n_out + n0 + l15];
        b.y = Wr[(k + koff + 1) * n_out + n0 + l15];
        acc = __builtin_amdgcn_wmma_f32_16x16x4_f32(false, a, false, b,
                                                    (short)0, acc, false, false);
    }

    // Epilogue: bias (+ ReLU), store per C/D VGPR layout.
    const int   col = n0 + l15;
    const float bv  = bias[col];
    #pragma unroll
    for (int r = 0; r < 8; ++r) {
        int   row = node0 + r + half * 8;
        float v   = acc[r] + bv;
        if (relu) v = fmaxf(v, 0.0f);
        out[(long)row * n_out + col] = v;
    }
}

// ---------------- launch ----------------

extern "C" void kernel_launch(void* const* d_in, const int* in_sizes, int n_in,
                              void* d_out, int out_size, void* d_ws, size_t ws_size,
                              hipStream_t stream) {
    const float* x   = (const float*)d_in[0];
    const int*   ei  = (const int*)d_in[1];   // [2, E] int32
    const float* W1l = (const float*)d_in[2];
    const float* b1  = (const float*)d_in[3];
    const float* W1r = (const float*)d_in[4];
    const float* W2l = (const float*)d_in[5];
    const float* b2  = (const float*)d_in[6];
    const float* W2r = (const float*)d_in[7];
    float*       out = (float*)d_out;

    const long N = N_NODES;
    const long E = (long)in_sizes[1] / 2;
    const int* src = ei;
    const int* dst = ei + E;

    // Workspace layout (floats): deg[N] | agg[N*128] | h[N*128]  (~103 MB)
    float* deg = (float*)d_ws;
    float* agg = deg + N;
    float* h   = agg + N * (long)N_FEAT;

    const long aggN = N * (long)N_FEAT;
    const int ZB = 256;

    // --- degrees + zero agg ---
    zero_f32_kernel<<<2048, ZB, 0, stream>>>(deg, N + aggN);
    degree_kernel<<<(int)((E + ZB - 1) / ZB), ZB, 0, stream>>>(dst, deg, E);
    invdeg_kernel<<<(int)((N + ZB - 1) / ZB), ZB, 0, stream>>>(deg, N);

    // --- layer 1: aggregate x, then h = relu(agg*inv@W1l + x@W1r + b1) ---
    {
        long threads = E * 32;
        scatter_add_kernel<<<(int)((threads + ZB - 1) / ZB), ZB, 0, stream>>>(x, src, dst, agg, E);
    }
    sage_gemm_wmma_kernel<<<(int)(N / 16), 8 * 32, 0, stream>>>(
        agg, deg, x, W1l, W1r, b1, h, /*n_out=*/128, /*relu=*/1);

    // --- layer 2: re-zero agg, aggregate h, out = agg*inv@W2l + h@W2r + b2 ---
    zero_f32_kernel<<<2048, ZB, 0, stream>>>(agg, aggN);
    {
        long threads = E * 32;
        scatter_add_kernel<<<(int)((threads + ZB - 1) / ZB), ZB, 0, stream>>>(h, src, dst, agg, E);
    }
    sage_gemm_wmma_kernel<<<(int)(N / 16), 4 * 32, 0, stream>>>(
        agg, deg, h, W2l, W2r, b2, out, /*n_out=*/64, /*relu=*/0);
}